// GCN_42047729827910
// MI455X (gfx1250) — compile-verified
//
#include <hip/hip_runtime.h>
#include <hip/hip_bf16.h>
#include <math.h>

typedef __attribute__((ext_vector_type(2))) float v2f;
typedef __attribute__((ext_vector_type(8))) float v8f;

// ---------------------------------------------------------------------------
// deg / dis
// ---------------------------------------------------------------------------
__global__ void gcn_init_deg(float* __restrict__ deg, int n) {
    int i = blockIdx.x * blockDim.x + threadIdx.x;
    if (i < n) deg[i] = 1.0f;                       // self-loop contribution
}

__global__ void gcn_deg_edges(const int* __restrict__ dst, float* __restrict__ deg, int E) {
    int e = blockIdx.x * blockDim.x + threadIdx.x;
    if (e < E) atomicAdd(&deg[dst[e]], 1.0f);
}

__global__ void gcn_rsqrt(float* __restrict__ deg, int n) {
    int i = blockIdx.x * blockDim.x + threadIdx.x;
    if (i < n) deg[i] = rsqrtf(fmaxf(deg[i], 1.0f));   // deg -> dis in place
}

// ---------------------------------------------------------------------------
// Layer 1 GEMM via WMMA:  hw = x @ W1  (K=128, Nout=4 padded to 16)
// Also writes agg = hw*dis^2 + b1  (self-loop + bias pre-accumulated).
// One wave per 16-row tile; n is an exact multiple of 16 (500000 = 31250*16).
// B fragments (pre-masked, per-lane) are staged in LDS once per block, so the
// hot loop is: global_load_b64 (x) + ds_load_b64 (B) + v_wmma.
// ---------------------------------------------------------------------------
__global__ void gcn_gemm1_wmma(const float* __restrict__ x,
                               const float* __restrict__ W1,
                               const float* __restrict__ b1,
                               const float* __restrict__ dis,
                               float* __restrict__ hw,
                               float* __restrict__ agg,
                               int nTiles) {
    __shared__ float2 bfrag[32][32];                // [k-chunk][lane] = (b.x, b.y), 8KB

    // Cooperative fill of all 32 B fragments (zero-padded columns baked in).
    for (int slot = threadIdx.x; slot < 32 * 32; slot += blockDim.x) {
        int chunk = slot >> 5;
        int l     = slot & 31;
        int hf    = l >> 4;                         // K sub-pair selector
        int nb    = l & 15;                         // output column (pad >= 4)
        int nbc   = (nb < 4) ? nb : 0;
        float msk = (nb < 4) ? 1.0f : 0.0f;
        int kb    = chunk * 4 + 2 * hf;
        float2 bv;
        bv.x = W1[(size_t)kb * 4 + nbc] * msk;
        bv.y = W1[(size_t)(kb + 1) * 4 + nbc] * msk;
        bfrag[chunk][l] = bv;
    }
    __syncthreads();                                // all waves participate, then tail waves exit

    int wave = blockIdx.x * (blockDim.x >> 5) + (threadIdx.x >> 5);
    if (wave >= nTiles) return;                     // wave-uniform: EXEC stays full
    int lane = threadIdx.x & 31;
    int half = lane >> 4;                           // 0: K 0..1 of chunk, 1: K 2..3
    int l16  = lane & 15;

    int rowA = wave * 16 + l16;                     // A-matrix row held by this lane
    const float2* xr2 = (const float2*)(x + (size_t)rowA * 128);

    v8f c = {};
    #pragma unroll
    for (int k = 0; k < 128; k += 4) {
        // A fragment: cols (k + 2*half, k + 2*half + 1) of row rowA
        float2 av = xr2[(k >> 1) + half];
        v2f a; a.x = av.x; a.y = av.y;
        // B fragment from LDS (already masked)
        float2 bv = bfrag[k >> 2][lane];
        v2f b; b.x = bv.x; b.y = bv.y;
        c = __builtin_amdgcn_wmma_f32_16x16x4_f32(false, a, false, b,
                                                  (short)0, c, false, false);
    }

    // D layout: VGPR r -> M = r + 8*half, N = l16.  Only N < 4 is real output.
    int nB = l16;
    if (nB < 4) {
        #pragma unroll
        for (int r = 0; r < 8; ++r) {
            int row = wave * 16 + r + 8 * half;
            float v  = c[r];
            float d  = dis[row];
            float d2 = d * d;
            hw [(size_t)row * 4 + nB] = v;
            agg[(size_t)row * 4 + nB] = v * d2 + b1[nB];
        }
    }
}

// ---------------------------------------------------------------------------
// Tiny dense layers (FIN<=4) fused with agg init
// ---------------------------------------------------------------------------
template <int FIN, int FOUT>
__global__ void gcn_small_gemm_init(const float* __restrict__ h,
                                    const float* __restrict__ W,
                                    const float* __restrict__ bb,
                                    const float* __restrict__ dis,
                                    float* __restrict__ hw,
                                    float* __restrict__ agg, int n) {
    int i = blockIdx.x * blockDim.x + threadIdx.x;
    if (i >= n) return;
    float hv[FIN];
    #pragma unroll
    for (int f = 0; f < FIN; ++f) hv[f] = h[(size_t)i * FIN + f];
    float d = dis[i];
    float d2 = d * d;
    #pragma unroll
    for (int o = 0; o < FOUT; ++o) {
        float s = 0.0f;
        #pragma unroll
        for (int f = 0; f < FIN; ++f) s += hv[f] * W[f * FOUT + o];
        hw [(size_t)i * FOUT + o] = s;
        agg[(size_t)i * FOUT + o] = s * d2 + bb[o];
    }
}

// ---------------------------------------------------------------------------
// Edge aggregation: agg[dst] += dis[src]*dis[dst] * hw[src]   (L2-resident)
// ---------------------------------------------------------------------------
template <int F>
__global__ void gcn_edge_agg(const int* __restrict__ src,
                             const int* __restrict__ dst,
                             const float* __restrict__ dis,
                             const float* __restrict__ hw,
                             float* __restrict__ agg, int E) {
    int e = blockIdx.x * blockDim.x + threadIdx.x;
    if (e >= E) return;
    int s = src[e];
    int d = dst[e];
    float w = dis[s] * dis[d];
    float m[F];
    if (F == 4) {
        float4 t = *(const float4*)(hw + (size_t)s * 4);
        m[0] = t.x; m[1] = t.y; m[2] = t.z; m[3] = t.w;
    } else {
        float2 t = *(const float2*)(hw + (size_t)s * 2);
        m[0] = t.x; m[1] = t.y;
    }
    #pragma unroll
    for (int f = 0; f < F; ++f)
        atomicAdd(&agg[(size_t)d * F + f], m[f] * w);
}

// ---------------------------------------------------------------------------
// tanh finalize (elementwise over n*F)
// ---------------------------------------------------------------------------
__global__ void gcn_tanh(const float* __restrict__ agg, float* __restrict__ h, int total) {
    int i = blockIdx.x * blockDim.x + threadIdx.x;
    if (i < total) h[i] = tanhf(agg[i]);
}

// ---------------------------------------------------------------------------
// Classifier: out = h3 @ Wc + bc   (2 -> 4)
// ---------------------------------------------------------------------------
__global__ void gcn_classifier(const float* __restrict__ h3,
                               const float* __restrict__ Wc,
                               const float* __restrict__ bc,
                               float* __restrict__ out, int n) {
    int i = blockIdx.x * blockDim.x + threadIdx.x;
    if (i >= n) return;
    float2 hv = *(const float2*)(h3 + (size_t)i * 2);
    #pragma unroll
    for (int c = 0; c < 4; ++c)
        out[(size_t)i * 4 + c] = hv.x * Wc[c] + hv.y * Wc[4 + c] + bc[c];
}

// ---------------------------------------------------------------------------
extern "C" void kernel_launch(void* const* d_in, const int* in_sizes, int n_in,
                              void* d_out, int out_size, void* d_ws, size_t ws_size,
                              hipStream_t stream) {
    const float* x  = (const float*)d_in[0];
    const int*   ei = (const int*)d_in[1];
    const float* W1 = (const float*)d_in[2];
    const float* b1 = (const float*)d_in[3];
    const float* W2 = (const float*)d_in[4];
    const float* b2 = (const float*)d_in[5];
    const float* W3 = (const float*)d_in[6];
    const float* b3 = (const float*)d_in[7];
    const float* Wc = (const float*)d_in[8];
    const float* bc = (const float*)d_in[9];

    int n = in_sizes[0] / 128;          // 500000
    int E = in_sizes[1] / 2;            // 16000000
    const int* src = ei;
    const int* dst = ei + E;

    float* out = (float*)d_out;             // [n*4]
    float* h3  = out + (size_t)n * 4;       // [n*2] (second tuple output)

    float* wsf = (float*)d_ws;
    float* dis = wsf;                       // n
    float* hw  = dis + n;                   // 4n
    float* agg = hw + (size_t)n * 4;        // 4n
    float* h   = agg + (size_t)n * 4;       // 4n

    const int B  = 256;
    int gN  = (n + B - 1) / B;
    int gE  = (E + B - 1) / B;
    int g4N = (n * 4 + B - 1) / B;
    int g2N = (n * 2 + B - 1) / B;

    // degree -> dis
    gcn_init_deg<<<gN, B, 0, stream>>>(dis, n);
    gcn_deg_edges<<<gE, B, 0, stream>>>(dst, dis, E);
    gcn_rsqrt<<<gN, B, 0, stream>>>(dis, n);

    // layer 1 (WMMA GEMM, K=128)
    int nTiles = n / 16;                    // exact: 31250
    int wpb = B / 32;
    gcn_gemm1_wmma<<<(nTiles + wpb - 1) / wpb, B, 0, stream>>>(x, W1, b1, dis, hw, agg, nTiles);
    gcn_edge_agg<4><<<gE, B, 0, stream>>>(src, dst, dis, hw, agg, E);
    gcn_tanh<<<g4N, B, 0, stream>>>(agg, h, n * 4);

    // layer 2 (4 -> 4)
    gcn_small_gemm_init<4, 4><<<gN, B, 0, stream>>>(h, W2, b2, dis, hw, agg, n);
    gcn_edge_agg<4><<<gE, B, 0, stream>>>(src, dst, dis, hw, agg, E);
    gcn_tanh<<<g4N, B, 0, stream>>>(agg, h, n * 4);

    // layer 3 (4 -> 2) -> h3 straight into d_out second region
    gcn_small_gemm_init<4, 2><<<gN, B, 0, stream>>>(h, W3, b3, dis, hw, agg, n);
    gcn_edge_agg<2><<<gE, B, 0, stream>>>(src, dst, dis, hw, agg, E);
    gcn_tanh<<<g2N, B, 0, stream>>>(agg, h3, n * 2);

    // classifier (2 -> 4)
    gcn_classifier<<<gN, B, 0, stream>>>(h3, Wc, bc, out, n);
}